// DeformConv_d_25512105738555
// MI455X (gfx1250) — compile-verified
//
#include <hip/hip_runtime.h>

typedef float v2f __attribute__((ext_vector_type(2)));
typedef float v8f __attribute__((ext_vector_type(8)));

#define B_   2
#define C_   16
#define T_   8
#define H_   64
#define W_   64
#define CO_  16
#define K_   27

#define X_CSTR   (T_*H_*W_)        /* 32768 */
#define X_BSTR   (C_*X_CSTR)       /* 524288 */
#define TMP_CSTR (T_*H_*W_)        /* 32768 */
#define TMP_BSTR (2*K_*TMP_CSTR)
#define O_CSTR   (T_*H_*W_)
#define O_BSTR   (CO_*O_CSTR)

// One wave (32 lanes) computes a 16(co) x 16(w-positions) output tile via
// V_WMMA_F32_16X16X4_F32 over the 432-deep (k,c) contraction in fp32.
__global__ __launch_bounds__(256) void deform_conv3d_wmma(
    const float* __restrict__ x, const float* __restrict__ temp,
    const float* __restrict__ weight, const float* __restrict__ bias,
    float* __restrict__ out)
{
    // Stage weights into LDS reorganized as [k][c][co] so A-fragments are
    // contiguous-stride ds loads. 27*16*16 floats = 27648 B (fits easily).
    __shared__ float lds_w[K_ * C_ * CO_];
    for (int i = threadIdx.x; i < K_ * C_ * CO_; i += 256) {
        int co = i / (C_ * K_);
        int r  = i - co * (C_ * K_);
        int c  = r / K_;
        int k  = r - c * K_;
        lds_w[(k * C_ + c) * CO_ + co] = weight[i];
    }
    __syncthreads();

    const int lane = threadIdx.x & 31;
    const int half = lane >> 4;       // 0: lanes 0-15, 1: lanes 16-31
    const int n    = lane & 15;       // output position within tile / co row

    // tile id = global wave id; tiles = B*T*H*(W/16) = 4096
    const int wave = blockIdx.x * 8 + (threadIdx.x >> 5);
    const int wt   = wave & 3;        // w-tile
    int rest       = wave >> 2;
    const int h    = rest & 63;
    rest >>= 6;
    const int t    = rest & 7;
    const int b    = rest >> 3;

    const int wout = wt * 16 + n;

    // D/C layout: VGPR v holds co = v + 8*half, col n. Init with bias.
    v8f acc;
#pragma unroll
    for (int v = 0; v < 8; ++v) acc[v] = bias[v + 8 * half];

    const float* tmpB = temp + b * TMP_BSTR + (t * H_ + h) * W_ + wout;
    const float* xB   = x + b * X_BSTR;
    const float* lwB  = &lds_w[n];    // co = lane & 15

    for (int kt = 0; kt < 3; ++kt) {
        const int ti = t - 1 + kt;            // wave-uniform
        if ((unsigned)ti >= (unsigned)T_) continue;  // EXEC stays all-1s
        const float* xT = xB + ti * (H_ * W_);
        for (int kh = 0; kh < 3; ++kh) {
            for (int kw = 0; kw < 3; ++kw) {
                const int k = (kt * 3 + kh) * 3 + kw;
                // learned offsets for this tap at this output position
                const float dh = tmpB[(2 * k)     * TMP_CSTR];
                const float dw = tmpB[(2 * k + 1) * TMP_CSTR];
                const float ph = (float)(h - 1 + kh) + dh;
                const float pw = (float)(wout - 1 + kw) + dw;
                const float h0f = floorf(ph);
                const float w0f = floorf(pw);
                const float fh = ph - h0f;
                const float fw = pw - w0f;
                const int h0 = (int)h0f, w0 = (int)w0f;
                const int h1 = h0 + 1,  w1 = w0 + 1;
                const float vh0 = (h0 >= 0 && h0 < H_) ? 1.0f : 0.0f;
                const float vh1 = (h1 >= 0 && h1 < H_) ? 1.0f : 0.0f;
                const float vw0 = (w0 >= 0 && w0 < W_) ? 1.0f : 0.0f;
                const float vw1 = (w1 >= 0 && w1 < W_) ? 1.0f : 0.0f;
                const int h0c = min(max(h0, 0), H_ - 1);
                const int h1c = min(max(h1, 0), H_ - 1);
                const int w0c = min(max(w0, 0), W_ - 1);
                const int w1c = min(max(w1, 0), W_ - 1);
                const float c00 = (1.0f - fh) * (1.0f - fw) * vh0 * vw0;
                const float c01 = (1.0f - fh) * fw          * vh0 * vw1;
                const float c10 = fh          * (1.0f - fw) * vh1 * vw0;
                const float c11 = fh          * fw          * vh1 * vw1;
                const int i00 = h0c * W_ + w0c;
                const int i01 = h0c * W_ + w1c;
                const int i10 = h1c * W_ + w0c;
                const int i11 = h1c * W_ + w1c;

                const float* lw = lwB + k * (C_ * CO_);
#pragma unroll
                for (int c0 = 0; c0 < 16; c0 += 4) {
                    // A 16x4 f32: VGPR0 = K0|K2, VGPR1 = K1|K3 across halves
                    // B 4x16 f32: same half split -> channels c0+2*half, +1
                    const int ca = c0 + 2 * half;
                    v2f a, bm;
                    a.x = lw[ ca      * CO_];
                    a.y = lw[(ca + 1) * CO_];
                    const float* xa = xT + ca * X_CSTR;
                    const float* xb = xa + X_CSTR;
                    bm.x = c00 * xa[i00] + c01 * xa[i01]
                         + c10 * xa[i10] + c11 * xa[i11];
                    bm.y = c00 * xb[i00] + c01 * xb[i01]
                         + c10 * xb[i10] + c11 * xb[i11];
                    acc = __builtin_amdgcn_wmma_f32_16x16x4_f32(
                              false, a, false, bm, (short)0, acc, false, false);
                }
            }
        }
    }

    // out[b][co][t][h][wout], co = v + 8*half
    float* oB = out + b * O_BSTR + (t * H_ + h) * W_ + wout;
#pragma unroll
    for (int v = 0; v < 8; ++v) {
        oB[(v + 8 * half) * O_CSTR] = acc[v];
    }
}

extern "C" void kernel_launch(void* const* d_in, const int* in_sizes, int n_in,
                              void* d_out, int out_size, void* d_ws, size_t ws_size,
                              hipStream_t stream) {
    const float* x      = (const float*)d_in[0];
    const float* temp   = (const float*)d_in[1];
    const float* weight = (const float*)d_in[2];
    const float* bias   = (const float*)d_in[3];
    float* out          = (float*)d_out;

    // tiles = B*T*H*(W/16) = 2*8*64*4 = 4096 waves; 8 waves/block -> 512 blocks
    dim3 grid(512), block(256);
    hipLaunchKernelGGL(deform_conv3d_wmma, grid, block, 0, stream,
                       x, temp, weight, bias, out);
}